// MultiHeadDotProduct_36146444763806
// MI455X (gfx1250) — compile-verified
//
#include <hip/hip_runtime.h>
#include <hip/hip_bf16.h>

typedef __attribute__((ext_vector_type(2))) float v2f;
typedef __attribute__((ext_vector_type(8))) float v8f;

#define NHEAD   8
#define HDIM    16
#define QDIM    128
#define NNODES  20000
#define NEDGES  640000
#define ROWTILES (NNODES / 16)   // 1250

// ---------------------------------------------------------------------------
// Kernel 1: fused q/k/v projections.  Y = X @ W^T + b  via v_wmma_f32_16x16x4
// One wave computes a 16-row x 128-col slab of one of the three outputs.
// A (16x4, f32) lane layout: lane L -> M = L%16, VGPR0/1 = K = 2*(L/16)+{0,1}
// B (4x16) mirrors it with N = L%16; since Y = X @ W^T, B[k][n] = W[n][k].
// C/D (16x16): VGPR r -> row r + 8*(L/16), col L%16.
// ---------------------------------------------------------------------------
__global__ __launch_bounds__(256)
void proj_kernel(const float* __restrict__ q, const float* __restrict__ k,
                 const float* __restrict__ v,
                 const float* __restrict__ Wq, const float* __restrict__ bq,
                 const float* __restrict__ Wk, const float* __restrict__ bk,
                 const float* __restrict__ Wv, const float* __restrict__ bv,
                 float* __restrict__ qh, float* __restrict__ kh,
                 float* __restrict__ vh)
{
    int wave = (int)((blockIdx.x * blockDim.x + threadIdx.x) >> 5);
    if (wave >= 3 * ROWTILES) return;
    int lane = (int)(threadIdx.x & 31);

    int mat     = wave / ROWTILES;     // 0=q, 1=k, 2=v
    int rowTile = wave % ROWTILES;

    const float *X, *W, *B;
    float *Y;
    if (mat == 0)      { X = q; W = Wq; B = bq; Y = qh; }
    else if (mat == 1) { X = k; W = Wk; B = bk; Y = kh; }
    else               { X = v; W = Wv; B = bv; Y = vh; }

    int mrow  = rowTile * 16 + (lane & 15);   // A row / B column index
    int khalf = (lane >> 4) * 2;              // K offset 0 or 2 within a K=4 step

    v8f acc[8] = {};                          // 8 col-tiles of 16x16 f32

    const float* xrow = X + (size_t)mrow * QDIM;
    for (int kb = 0; kb < QDIM; kb += 4) {
        v2f a = *(const v2f*)(xrow + kb + khalf);
#pragma unroll
        for (int t = 0; t < 8; ++t) {
            int col = t * 16 + (lane & 15);
            v2f b = *(const v2f*)(W + (size_t)col * QDIM + kb + khalf);
            acc[t] = __builtin_amdgcn_wmma_f32_16x16x4_f32(
                         false, a, false, b, (short)0, acc[t], false, false);
        }
    }

    int rowOff = 8 * (lane >> 4);
#pragma unroll
    for (int t = 0; t < 8; ++t) {
        int   col  = t * 16 + (lane & 15);
        float bias = B[col];
        float* yb  = Y + (size_t)(rowTile * 16 + rowOff) * QDIM + col;
#pragma unroll
        for (int r = 0; r < 8; ++r)
            yb[(size_t)r * QDIM] = acc[t][r] + bias;
    }
}

// ---------------------------------------------------------------------------
// Kernel 2: per-edge scores.  One wave per edge; lane L covers dims [4L,4L+4)
// (always inside head h = L/4).  4-lane shuffle reduce -> score per head,
// ex = exp(score/4); lane 4h stores ex and atomically accumulates denom.
// qh/kh (20.5 MB total) stay resident in the 192 MB L2, so the random
// gathers are L2-bandwidth bound, not HBM bound.
// ---------------------------------------------------------------------------
__global__ __launch_bounds__(256)
void edge_score_kernel(const float* __restrict__ qh, const float* __restrict__ kh,
                       const int* __restrict__ self_idx,
                       const int* __restrict__ nbr_idx,
                       float* __restrict__ exbuf, float* __restrict__ denom)
{
    int e = (int)((blockIdx.x * blockDim.x + threadIdx.x) >> 5);
    if (e >= NEDGES) return;
    int lane = (int)(threadIdx.x & 31);

    int si = self_idx[e];
    int ni = nbr_idx[e];

    float4 a = *(const float4*)(qh + (size_t)si * QDIM + lane * 4);
    float4 b = *(const float4*)(kh + (size_t)ni * QDIM + lane * 4);
    float p = a.x * b.x + a.y * b.y + a.z * b.z + a.w * b.w;
    p += __shfl_xor(p, 1, 32);
    p += __shfl_xor(p, 2, 32);           // lanes 4h..4h+3 all hold score[h]

    float ex = __expf(p * 0.25f);        // 1/sqrt(HDIM) = 0.25
    if ((lane & 3) == 0) {
        int h = lane >> 2;
        exbuf[(size_t)e * NHEAD + h] = ex;
        atomicAdd(&denom[(size_t)si * NHEAD + h], ex);
    }
}

// ---------------------------------------------------------------------------
// Kernel 3: attnsum[i,h] = sum_e ex[e,h]/denom[i,h] over edges with self==i.
// Because the reference gathers v by self_indices, the per-node output is
// vh[i,h,:] * attnsum[i,h] -- 8 scalars per edge instead of 128 floats.
// ---------------------------------------------------------------------------
__global__ __launch_bounds__(256)
void attn_sum_kernel(const float* __restrict__ exbuf,
                     const int* __restrict__ self_idx,
                     const float* __restrict__ denom,
                     float* __restrict__ attnsum)
{
    long long t = (long long)blockIdx.x * blockDim.x + threadIdx.x;
    if (t >= (long long)NEDGES * NHEAD) return;
    int e = (int)(t >> 3);
    int h = (int)(t & 7);
    int si = self_idx[e];
    float attn = exbuf[t] / denom[(size_t)si * NHEAD + h];
    atomicAdd(&attnsum[(size_t)si * NHEAD + h], attn);
}

// ---------------------------------------------------------------------------
// Kernel 4: out = (vh * attnsum_per_head) @ Wo^T + bo, scale fused into the
// WMMA A-operand load (both elements of a K-pair share one head).
// ---------------------------------------------------------------------------
__global__ __launch_bounds__(256)
void out_gemm_kernel(const float* __restrict__ vh,
                     const float* __restrict__ attnsum,
                     const float* __restrict__ Wo, const float* __restrict__ bo,
                     float* __restrict__ out)
{
    int wave = (int)((blockIdx.x * blockDim.x + threadIdx.x) >> 5);
    if (wave >= ROWTILES) return;
    int lane = (int)(threadIdx.x & 31);

    int mrow  = wave * 16 + (lane & 15);
    int khalf = (lane >> 4) * 2;

    v8f acc[8] = {};

    const float* vrow = vh + (size_t)mrow * QDIM;
    const float* srow = attnsum + (size_t)mrow * NHEAD;
    for (int kb = 0; kb < QDIM; kb += 4) {
        float s = srow[(kb + khalf) >> 4];      // head scale for this K-pair
        v2f a = *(const v2f*)(vrow + kb + khalf);
        a[0] *= s;
        a[1] *= s;
#pragma unroll
        for (int t = 0; t < 8; ++t) {
            int col = t * 16 + (lane & 15);
            v2f b = *(const v2f*)(Wo + (size_t)col * QDIM + kb + khalf);
            acc[t] = __builtin_amdgcn_wmma_f32_16x16x4_f32(
                         false, a, false, b, (short)0, acc[t], false, false);
        }
    }

    int rowOff = 8 * (lane >> 4);
#pragma unroll
    for (int t = 0; t < 8; ++t) {
        int   col  = t * 16 + (lane & 15);
        float bias = bo[col];
        float* yb  = out + (size_t)(wave * 16 + rowOff) * QDIM + col;
#pragma unroll
        for (int r = 0; r < 8; ++r)
            yb[(size_t)r * QDIM] = acc[t][r] + bias;
    }
}

// ---------------------------------------------------------------------------
extern "C" void kernel_launch(void* const* d_in, const int* in_sizes, int n_in,
                              void* d_out, int out_size, void* d_ws, size_t ws_size,
                              hipStream_t stream)
{
    const float* q   = (const float*)d_in[0];
    const float* k   = (const float*)d_in[1];
    const float* v   = (const float*)d_in[2];
    const int*   self_idx = (const int*)d_in[3];
    const int*   nbr_idx  = (const int*)d_in[4];
    const float* Wq  = (const float*)d_in[5];
    const float* bq  = (const float*)d_in[6];
    const float* Wk  = (const float*)d_in[7];
    const float* bk  = (const float*)d_in[8];
    const float* Wv  = (const float*)d_in[9];
    const float* bv  = (const float*)d_in[10];
    const float* Wo  = (const float*)d_in[11];
    const float* bo  = (const float*)d_in[12];
    float* out = (float*)d_out;

    // Workspace layout (floats): qh | kh | vh | exbuf | denom | attnsum
    float* ws      = (float*)d_ws;
    float* qh      = ws;                                 // 20000*128
    float* kh      = qh + (size_t)NNODES * QDIM;
    float* vh      = kh + (size_t)NNODES * QDIM;
    float* exbuf   = vh + (size_t)NNODES * QDIM;         // 640000*8
    float* denom   = exbuf + (size_t)NEDGES * NHEAD;     // 20000*8
    float* attnsum = denom + (size_t)NNODES * NHEAD;     // 20000*8

    // Zero the accumulation buffers (graph-capture-safe).
    hipMemsetAsync(denom, 0, 2 * (size_t)NNODES * NHEAD * sizeof(float), stream);

    // 1) projections: 3*1250 waves, 8 waves (256 threads) per block
    {
        int waves  = 3 * ROWTILES;
        int blocks = (waves + 7) / 8;
        proj_kernel<<<blocks, 256, 0, stream>>>(q, k, v, Wq, bq, Wk, bk, Wv, bv,
                                                qh, kh, vh);
    }
    // 2) edge scores: one wave per edge
    {
        int blocks = (NEDGES + 7) / 8;
        edge_score_kernel<<<blocks, 256, 0, stream>>>(qh, kh, self_idx, nbr_idx,
                                                      exbuf, denom);
    }
    // 3) per-node attention sums: one thread per (edge, head)
    {
        long long threads = (long long)NEDGES * NHEAD;
        int blocks = (int)((threads + 255) / 256);
        attn_sum_kernel<<<blocks, 256, 0, stream>>>(exbuf, self_idx, denom,
                                                    attnsum);
    }
    // 4) output projection
    {
        int blocks = (ROWTILES + 7) / 8;
        out_gemm_kernel<<<blocks, 256, 0, stream>>>(vh, attnsum, Wo, bo, out);
    }
    (void)in_sizes; (void)n_in; (void)out_size; (void)ws_size;
}